// BNNFC_395136991534
// MI455X (gfx1250) — compile-verified
//
#include <hip/hip_runtime.h>

typedef __attribute__((ext_vector_type(2))) float v2f;
typedef __attribute__((ext_vector_type(8))) float v8f;

#define B_    64
#define T_    200
#define IN_   512
#define HID_  1024
#define OUT_  512
#define RING_ 21      // DELAY(20) + 1 slots: read slot (t+1)%21, write slot t%21
#define R_MEM_ 0.1f

__device__ __forceinline__ float sigmoidf_(float x) {
    return 1.0f / (1.0f + __expf(-x));
}

__global__ void glifr_zero(float* __restrict__ p, int n) {
    int i = blockIdx.x * blockDim.x + threadIdx.x;
    int stride = gridDim.x * blockDim.x;
    for (; i < n; i += stride) p[i] = 0.0f;
}

// LDS-tiled transpose: out[c][r] = in[r][c]. rows/cols are multiples of 32.
// Run once per launch so the step kernel's B-fragments become contiguous b64
// loads along K.
__global__ __launch_bounds__(256)
void glifr_transpose(const float* __restrict__ in, float* __restrict__ outp,
                     int rows, int cols)
{
    __shared__ float tile[32][33];
    const int c0 = blockIdx.x * 32;
    const int r0 = blockIdx.y * 32;
    const int tx = threadIdx.x & 31;
    const int ty = threadIdx.x >> 5;      // 0..7
    #pragma unroll
    for (int j = 0; j < 32; j += 8)
        tile[ty + j][tx] = in[(size_t)(r0 + ty + j) * cols + (c0 + tx)];
    __syncthreads();
    #pragma unroll
    for (int j = 0; j < 32; j += 8)
        outp[(size_t)(c0 + ty + j) * rows + (r0 + tx)] = tile[tx][ty + j];
}

// One wave (wave32) owns one 16x16 tile of syn[B,H]; fused GLIFR state update
// in the WMMA epilogue. grid: 256 tiles = 64 blocks x 4 waves.
__global__ __launch_bounds__(128)
void glifr_step(const float* __restrict__ x,
                const float* __restrict__ WinT,    // [HID][IN]
                const float* __restrict__ WlatT,   // [HID][HID]
                const float* __restrict__ thresh,
                const float* __restrict__ tkm,
                const float* __restrict__ task,
                const float* __restrict__ aamp,
                const float* __restrict__ tasr,
                float* __restrict__ ring,
                float* __restrict__ firing,
                float* __restrict__ volt,
                float* __restrict__ asc,
                int t)
{
    const int lane = threadIdx.x & 31;
    const int wave = threadIdx.x >> 5;
    const int tile = blockIdx.x * 4 + wave;   // 0..255
    const int bi = tile >> 6;                 // batch tile 0..3
    const int hj = tile & 63;                 // hidden tile 0..63
    const int m0 = bi * 16;
    const int n0 = hj * 16;

    const int lm = lane & 15;                 // A row / B col within tile
    const int lk = (lane >> 4) << 1;          // K sub-offset {0,2}
    const int n  = n0 + lm;                   // hidden column this lane owns

    v8f acc = {};

    // ---- input projection: x[b, t, :] @ W_in (K = 512) ----
    const float* Arow  = x + (size_t)(m0 + lm) * (T_ * IN_) + (size_t)t * IN_;
    const float* BrowI = WinT + (size_t)n * IN_;          // W_in^T[n][k]
    #pragma unroll 4
    for (int kk = 0; kk < IN_; kk += 4) {
        v2f a = *(const v2f*)(Arow  + kk + lk);           // contiguous b64
        v2f b = *(const v2f*)(BrowI + kk + lk);           // contiguous b64
        acc = __builtin_amdgcn_wmma_f32_16x16x4_f32(false, a, false, b,
                                                    (short)0, acc, false, false);
    }

    // ---- recurrent: firing_{t-20} @ W_lat (K = 1024) ----
    const int rs = (t + 1) % RING_;                       // == (t - 20) mod 21
    const float* Frow  = ring + (size_t)rs * (B_ * HID_) + (size_t)(m0 + lm) * HID_;
    const float* BrowL = WlatT + (size_t)n * HID_;        // W_lat^T[n][k]
    #pragma unroll 4
    for (int kk = 0; kk < HID_; kk += 4) {
        v2f a = *(const v2f*)(Frow  + kk + lk);
        v2f b = *(const v2f*)(BrowL + kk + lk);
        acc = __builtin_amdgcn_wmma_f32_16x16x4_f32(false, a, false, b,
                                                    (short)0, acc, false, false);
    }

    // ---- fused GLIFR epilogue (per-hidden params loaded once per lane) ----
    const float sm  = sigmoidf_(tkm[n]);                  // DT * k_m
    const float th  = thresh[n];
    const float sa0 = sigmoidf_(task[n]);                 // DT * k_asc[0]
    const float sa1 = sigmoidf_(task[HID_ + n]);          // DT * k_asc[1]
    const float r0  = 1.0f - 2.0f * sigmoidf_(tasr[n]);
    const float r1  = 1.0f - 2.0f * sigmoidf_(tasr[HID_ + n]);
    const float am0 = aamp[n];
    const float am1 = aamp[HID_ + n];

    float* ringW = ring + (size_t)(t % RING_) * (B_ * HID_);
    const int mhi = (lane >> 4) << 3;                     // +8 for upper half-wave

    #pragma unroll
    for (int r = 0; r < 8; ++r) {
        const int m = m0 + r + mhi;                       // C/D layout: M = r (+8)
        const size_t idx = (size_t)m * HID_ + n;
        float f  = firing[idx];
        float v  = volt[idx];
        float a0 = asc[idx];
        float a1 = asc[(size_t)B_ * HID_ + idx];
        float a0n = (am0 + r0 * a0) * f * sa0 + (1.0f - sa0) * a0;
        float a1n = (am1 + r1 * a1) * f * sa1 + (1.0f - sa1) * a1;
        float vn  = (R_MEM_ * sm) * (acc[r] + a0n + a1n) + (1.0f - sm) * v;
        float fn  = sigmoidf_(vn - th);                   // SIGMA_V = 1
        firing[idx] = fn;
        volt[idx]   = vn;
        asc[idx]    = a0n;
        asc[(size_t)B_ * HID_ + idx] = a1n;
        ringW[idx]  = fn;
    }
}

// Per-step readout: out[b,t,:] = firing @ W_out^T + b_out.
// 128 tiles = 32 blocks x 4 waves. W_out is [OUT,HID] so the B-fragment is
// already a contiguous fp32 pair along K.
__global__ __launch_bounds__(128)
void glifr_out(const float* __restrict__ firing,
               const float* __restrict__ Wout,
               const float* __restrict__ bout,
               float* __restrict__ out,
               int t)
{
    const int lane = threadIdx.x & 31;
    const int wave = threadIdx.x >> 5;
    const int tile = blockIdx.x * 4 + wave;   // 0..127
    const int bi = tile >> 5;                 // 0..3
    const int oj = tile & 31;                 // 0..31
    const int m0 = bi * 16;
    const int n0 = oj * 16;
    const int lm = lane & 15;
    const int lk = (lane >> 4) << 1;
    const int n  = n0 + lm;

    v8f acc = {};
    const float* Arow = firing + (size_t)(m0 + lm) * HID_;
    const float* Brow = Wout + (size_t)n * HID_;          // W_out[n][k]
    #pragma unroll 4
    for (int kk = 0; kk < HID_; kk += 4) {
        v2f a = *(const v2f*)(Arow + kk + lk);
        v2f b = *(const v2f*)(Brow + kk + lk);
        acc = __builtin_amdgcn_wmma_f32_16x16x4_f32(false, a, false, b,
                                                    (short)0, acc, false, false);
    }

    const float bias = bout[n];
    const int mhi = (lane >> 4) << 3;
    #pragma unroll
    for (int r = 0; r < 8; ++r) {
        const int m = m0 + r + mhi;
        out[(size_t)m * (T_ * OUT_) + (size_t)t * OUT_ + n] = acc[r] + bias;
    }
}

extern "C" void kernel_launch(void* const* d_in, const int* in_sizes, int n_in,
                              void* d_out, int out_size, void* d_ws, size_t ws_size,
                              hipStream_t stream)
{
    const float* x    = (const float*)d_in[0];   // [B,T,IN]
    const float* Win  = (const float*)d_in[1];   // [IN,HID]
    const float* Wlat = (const float*)d_in[2];   // [HID,HID]
    const float* th   = (const float*)d_in[3];   // [1,HID]
    const float* tkm  = (const float*)d_in[4];   // [1,HID]
    const float* task = (const float*)d_in[5];   // [2,1,HID]
    const float* aamp = (const float*)d_in[6];   // [2,1,HID]
    const float* tasr = (const float*)d_in[7];   // [2,1,HID]
    const float* Wout = (const float*)d_in[8];   // [OUT,HID]
    const float* bout = (const float*)d_in[9];   // [OUT]
    float* out = (float*)d_out;                  // [B,T,OUT]

    float* ws     = (float*)d_ws;
    float* ring   = ws;                                   // RING_*B*H
    float* firing = ring   + (size_t)RING_ * B_ * HID_;   // B*H
    float* volt   = firing + (size_t)B_ * HID_;           // B*H
    float* asc    = volt   + (size_t)B_ * HID_;           // 2*B*H
    float* WinT   = asc    + (size_t)2 * B_ * HID_;       // HID*IN
    float* WlatT  = WinT   + (size_t)HID_ * IN_;          // HID*HID
    const int nstate = (RING_ + 4) * B_ * HID_;           // 25 * 64K floats

    glifr_zero<<<256, 256, 0, stream>>>(ws, nstate);

    // One-time weight transposes (fully overwrite their regions; no init needed).
    glifr_transpose<<<dim3(HID_ / 32, IN_ / 32), 256, 0, stream>>>(Win,  WinT,  IN_,  HID_);
    glifr_transpose<<<dim3(HID_ / 32, HID_ / 32), 256, 0, stream>>>(Wlat, WlatT, HID_, HID_);

    for (int t = 0; t < T_; ++t) {
        glifr_step<<<64, 128, 0, stream>>>(x, WinT, WlatT, th, tkm, task, aamp, tasr,
                                           ring, firing, volt, asc, t);
        glifr_out<<<32, 128, 0, stream>>>(firing, Wout, bout, out, t);
    }
}